// SAGP_38714835206189
// MI455X (gfx1250) — compile-verified
//
#include <hip/hip_runtime.h>

#define G_   64
#define NPG_ 1024
#define NN   (G_ * NPG_)      // 65536 nodes
#define EE   (NN * 16)        // 1048576 edges
#define EPG_ (EE / G_)        // 16384 edges per graph (graph-blocked by construction)
#define K1_  512
#define K2_  256
#define N2_  (G_ * K1_)       // 32768 pooled nodes (layer 1)
#define N3_  (G_ * K2_)       // 16384 pooled nodes (layer 2)

typedef __attribute__((ext_vector_type(2))) float v2f;
typedef __attribute__((ext_vector_type(8))) float v8f;
typedef __attribute__((ext_vector_type(4))) unsigned int v4u;
typedef __attribute__((ext_vector_type(8))) int v8i;
typedef __attribute__((ext_vector_type(4))) int v4i;

// ---------------------------------------------------------------------------
// Dense GEMM: Hout[nrows x 128] = X[nrows x 128] @ W[128 x 128], fp32 WMMA.
// 256 threads (8 waves). Each block does 128 rows; each wave a 16-row strip.
// W tile prefetched into LDS by the Tensor Data Mover, then deterministically
// (re)written by a coalesced copy so codegen sees real LDS stores (the TDM
// builtin is opaque to alias analysis) and semantics don't depend on the
// hand-packed D# encoding.
// ---------------------------------------------------------------------------
__global__ __launch_bounds__(256) void gemm128_wmma(const float* __restrict__ X,
                                                    const float* __restrict__ W,
                                                    float* __restrict__ Hout) {
  __shared__ float lw[128 * 128];   // sole shared array -> LDS offset 0
  int tid = threadIdx.x;

#if __has_builtin(__builtin_amdgcn_tensor_load_to_lds) && \
    __has_builtin(__builtin_amdgcn_s_wait_tensorcnt)
  if (tid < 32) {  // wave 0 issues the DMA (TDM is a per-wave op, EXEC ignored)
    unsigned long long wa = (unsigned long long)(uintptr_t)W;
    // D# group0: count=1 | lds_addr=0 | global_addr | type=2 ("image")
    v4u g0 = {1u, 0u, (unsigned)wa,
              (unsigned)((wa >> 32) & 0x01FFFFFFu) | (2u << 30)};
    // D# group1: mask=0, data_size=2(4B), tensor_dim0=128, tensor_dim1=128,
    //            tile_dim0=128, tile_dim1=128, tensor_dim0_stride=128
    v8i g1 = {(int)(2u << 16),      // data_size=4B
              (int)(128u << 16),    // tensor_dim0 lo16 (bits 63:48)
              (int)(128u << 16),    // tensor_dim1 lo16 (bits 95:80)
              (int)(128u << 16),    // tile_dim0 (bits 127:112)
              (int)128,             // tile_dim1 (bits 143:128)
              (int)128,             // tensor_dim0_stride lo32 (bits 191:160)
              0, 0};
    v4i gz4 = {0, 0, 0, 0};         // dims >2 unused (tile_dim2=0)
    v8i gz8 = {0, 0, 0, 0, 0, 0, 0, 0};
    __builtin_amdgcn_tensor_load_to_lds(g0, g1, gz4, gz4, gz8, 0);
    __builtin_amdgcn_s_wait_tensorcnt(0);
  }
  __syncthreads();  // TDM writes (if any) have landed before we overwrite
#endif
  // Deterministic staging (coalesced; W is L2-resident so this is cheap).
  for (int i = tid; i < 128 * 128; i += 256) lw[i] = W[i];
  __syncthreads();

  int wave = tid >> 5, lane = tid & 31;
  int row0 = blockIdx.x * 128 + wave * 16;
  int m = lane & 15;
  int half = lane >> 4;
  int kh = half * 2;
  const float* xr = X + (size_t)(row0 + m) * 128;

  v8f acc[8] = {};
  for (int k0 = 0; k0 < 128; k0 += 4) {
    v2f a;
    a.x = xr[k0 + kh];
    a.y = xr[k0 + kh + 1];
#pragma unroll
    for (int t = 0; t < 8; ++t) {
      int n0 = t * 16;
      v2f b;
      b.x = lw[(k0 + kh) * 128 + n0 + m];
      b.y = lw[(k0 + kh + 1) * 128 + n0 + m];
      acc[t] = __builtin_amdgcn_wmma_f32_16x16x4_f32(
          false, a, false, b, (short)0, acc[t], false, false);
    }
  }

  int rbase = row0 + half * 8;
#pragma unroll
  for (int t = 0; t < 8; ++t) {
    int col = t * 16 + m;
#pragma unroll
    for (int r = 0; r < 8; ++r)
      Hout[(size_t)(rbase + r) * 128 + col] = acc[t][r];
  }
}

// ---------------------------------------------------------------------------
// Fused degree + rsqrt per graph, all in LDS (replaces fill + global atomics).
// Block = NPGL threads, one block per graph.
// ---------------------------------------------------------------------------
template <int NPGL>
__global__ void deg_dinv(float* __restrict__ dinv_out,
                         const int* __restrict__ dst,
                         const float* __restrict__ ew) {
  __shared__ float deg[NPGL];
  int g = blockIdx.x, t = threadIdx.x, gb = g * NPGL;
  deg[t] = 1.0f;  // self loop
  __syncthreads();
  for (int e = g * EPG_ + t; e < (g + 1) * EPG_; e += NPGL) {
    float w = ew ? ew[e] : 1.0f;
    if (w != 0.0f) atomicAdd(&deg[dst[e] - gb], w);
  }
  __syncthreads();
  float v = deg[t];
  dinv_out[gb + t] = (v > 0.0f) ? rsqrtf(fmaxf(v, 1e-12f)) : 0.0f;
}

// ---------------------------------------------------------------------------
// Fused GCN aggregation: out = relu(segsum(norm*h[s]) + dinv^2*h + bias).
// One workgroup per (graph, 64-feature slice); accumulator tile lives in LDS
// (NPGL x 64 floats; 256KB max — fits 320KB/WGP). ds_add_f32 is bank-conflict
// free: lane owns features fbase+2*lane{,+1}. Edge gather = one b64 load.
// ---------------------------------------------------------------------------
template <int NPGL>
__global__ __launch_bounds__(1024) void agg_relu_lds(
    float* __restrict__ out, const float* __restrict__ h,
    const float* __restrict__ dinv, const float* __restrict__ bias,
    const int* __restrict__ src, const int* __restrict__ dst,
    const float* __restrict__ ew) {
  constexpr int FQ = 64;
  __shared__ float acc[NPGL * FQ];
  int g = blockIdx.x >> 1;
  int fbase = (blockIdx.x & 1) * FQ;
  int gb = g * NPGL;
  int t = threadIdx.x;
  for (int i = t; i < NPGL * FQ; i += 1024) acc[i] = 0.0f;
  __syncthreads();

  int wave = t >> 5, lane = t & 31;
  int f0 = fbase + 2 * lane;
  int e0 = g * EPG_ + wave * (EPG_ / 32);
  for (int i = 0; i < EPG_ / 32; ++i) {
    int e = e0 + i;
    int s = src[e], d = dst[e];
    float w = ew ? ew[e] : 1.0f;
    float coef = dinv[s] * dinv[d] * w;
    if (coef != 0.0f) {
      const float* hp = h + (size_t)s * 128 + f0;
      int base = (d - gb) * FQ + 2 * lane;
      atomicAdd(&acc[base],     coef * hp[0]);
      atomicAdd(&acc[base + 1], coef * hp[1]);
    }
  }
  __syncthreads();

  for (int i = t; i < NPGL * FQ; i += 1024) {
    int node = i >> 6;
    int f = fbase + (i & 63);
    float dv = dinv[gb + node];
    float v = acc[i] + dv * dv * h[(size_t)(gb + node) * 128 + f] + bias[f];
    out[(size_t)(gb + node) * 128 + f] = fmaxf(v, 0.0f);
  }
}

// ---------------------------------------------------------------------------
// Fused score conv + top-k per graph:
//  1) hs[node] = h[node,:] . Ws   (one wave per 32 nodes, shfl reduce)
//  2) LDS scalar edge aggregation (ds_add_f32)
//  3) finalize score (+bias, self term), write to global
//  4) descending bitonic sort; write perm/kept/nmap (kept fully written:
//     threads >= KK clear their node's flag -> no separate fill pass)
// ---------------------------------------------------------------------------
template <int NPGL, int KK>
__global__ void score_topk(float* __restrict__ score_out, int* __restrict__ perm,
                           int* __restrict__ kept, int* __restrict__ nmap,
                           const float* __restrict__ h,
                           const float* __restrict__ dinv,
                           const float* __restrict__ Ws,
                           const float* __restrict__ bs,
                           const int* __restrict__ src,
                           const int* __restrict__ dst,
                           const float* __restrict__ ew) {
  __shared__ float hs[NPGL];
  __shared__ float sc[NPGL];
  __shared__ int si[NPGL];
  int g = blockIdx.x, t = threadIdx.x, gb = g * NPGL;
  int wave = t >> 5, lane = t & 31;

  float w0 = Ws[lane], w1 = Ws[lane + 32], w2 = Ws[lane + 64], w3 = Ws[lane + 96];
  for (int i = 0; i < 32; ++i) {  // 32 nodes per wave (NPGL/32 waves)
    int node = wave * 32 + i;
    const float* hp = h + (size_t)(gb + node) * 128;
    float a = hp[lane] * w0 + hp[lane + 32] * w1 + hp[lane + 64] * w2 +
              hp[lane + 96] * w3;
    for (int o = 16; o > 0; o >>= 1) a += __shfl_down(a, o, 32);
    if (lane == 0) hs[node] = a;
  }
  sc[t] = 0.0f;
  __syncthreads();

  for (int e = g * EPG_ + t; e < (g + 1) * EPG_; e += NPGL) {
    int s = src[e], d = dst[e];
    float w = ew ? ew[e] : 1.0f;
    float coef = dinv[s] * dinv[d] * w;
    if (coef != 0.0f) atomicAdd(&sc[d - gb], coef * hs[s - gb]);
  }
  __syncthreads();

  {
    float dv = dinv[gb + t];
    float v = sc[t] + dv * dv * hs[t] + bs[0];
    sc[t] = v;
    score_out[gb + t] = v;
    si[t] = t;
  }
  __syncthreads();

  for (int k = 2; k <= NPGL; k <<= 1) {
    for (int j = k >> 1; j > 0; j >>= 1) {
      int ixj = t ^ j;
      if (ixj > t) {
        bool dir = ((t & k) == 0);
        if ((sc[t] < sc[ixj]) == dir) {
          float tf = sc[t]; sc[t] = sc[ixj]; sc[ixj] = tf;
          int ti = si[t]; si[t] = si[ixj]; si[ixj] = ti;
        }
      }
      __syncthreads();
    }
  }

  int node = gb + si[t];
  if (t < KK) {
    perm[g * KK + t] = node;
    kept[node] = 1;
    nmap[node] = g * KK + t;
  } else {
    kept[node] = 0;
  }
}

// xp[j,f] = h[perm[j],f] * tanh(score[perm[j]])
__global__ void gather_tanh(float* __restrict__ xp, const float* __restrict__ h,
                            const float* __restrict__ score,
                            const int* __restrict__ perm, int rows) {
  int idx = blockIdx.x * blockDim.x + threadIdx.x;
  if (idx >= rows * 128) return;
  int j = idx >> 7, f = idx & 127;
  int p = perm[j];
  xp[idx] = h[(size_t)p * 128 + f] * tanhf(score[p]);
}

// xo[g, 0:128]=max over KK rows, xo[g,128:256]=mean
template <int KK>
__global__ void readout_k(float* __restrict__ xo, const float* __restrict__ xp) {
  int g = blockIdx.x, f = threadIdx.x;  // 128 threads
  const float* b = xp + (size_t)g * KK * 128 + f;
  float m = -3.402823466e38f, s = 0.0f;
  for (int r = 0; r < KK; ++r) {
    float v = b[(size_t)r * 128];
    m = fmaxf(m, v);
    s += v;
  }
  xo[g * 256 + f] = m;
  xo[g * 256 + 128 + f] = s * (1.0f / (float)KK);
}

// edge relabel after pool 1 (layer-1 ew == 1 > 0 always)
__global__ void remap_edges(int* __restrict__ sn, int* __restrict__ dn,
                            float* __restrict__ ewn, const int* __restrict__ src,
                            const int* __restrict__ dst, const int* __restrict__ kept,
                            const int* __restrict__ nmap, int e) {
  int i = blockIdx.x * blockDim.x + threadIdx.x;
  if (i >= e) return;
  int s = src[i], d = dst[i];
  bool valid = kept[s] && kept[d];
  sn[i] = valid ? nmap[s] : 0;
  dn[i] = valid ? nmap[d] : 0;
  ewn[i] = valid ? 1.0f : 0.0f;
}

// final head: out[g] = relu((x1+x2) @ Wl1 + bl1) @ Wl2 + bl2
__global__ __launch_bounds__(128) void mlp_k(float* __restrict__ out,
                                             const float* __restrict__ x1,
                                             const float* __restrict__ x2,
                                             const float* __restrict__ Wl1,
                                             const float* __restrict__ bl1,
                                             const float* __restrict__ Wl2,
                                             const float* __restrict__ bl2) {
  __shared__ float z[256];
  __shared__ float red[128];
  int g = blockIdx.x, t = threadIdx.x;
  z[t] = x1[g * 256 + t] + x2[g * 256 + t];
  z[t + 128] = x1[g * 256 + 128 + t] + x2[g * 256 + 128 + t];
  __syncthreads();
  float acc = bl1[t];
  for (int j = 0; j < 256; ++j) acc += z[j] * Wl1[j * 128 + t];
  acc = fmaxf(acc, 0.0f);
  red[t] = acc * Wl2[t];
  __syncthreads();
  for (int s = 64; s > 0; s >>= 1) {
    if (t < s) red[t] += red[t + s];
    __syncthreads();
  }
  if (t == 0) out[g] = red[0] + bl2[0];
}

// ---------------------------------------------------------------------------
static inline void* wsalloc(char*& cur, size_t bytes) {
  void* p = (void*)cur;
  cur += (bytes + 255) & ~(size_t)255;
  return p;
}
static inline int cdiv(int a, int b) { return (a + b - 1) / b; }

extern "C" void kernel_launch(void* const* d_in, const int* in_sizes, int n_in,
                              void* d_out, int out_size, void* d_ws, size_t ws_size,
                              hipStream_t stream) {
  (void)in_sizes; (void)n_in; (void)out_size; (void)ws_size;
  const float* x   = (const float*)d_in[0];
  const int*   src = (const int*)  d_in[1];
  const int*   dst = (const int*)  d_in[2];
  const float* W1  = (const float*)d_in[3];
  const float* b1  = (const float*)d_in[4];
  const float* Ws1 = (const float*)d_in[5];
  const float* bs1 = (const float*)d_in[6];
  const float* W2  = (const float*)d_in[7];
  const float* b2  = (const float*)d_in[8];
  const float* Ws2 = (const float*)d_in[9];
  const float* bs2 = (const float*)d_in[10];
  const float* Wl1 = (const float*)d_in[11];
  const float* bl1 = (const float*)d_in[12];
  const float* Wl2 = (const float*)d_in[13];
  const float* bl2 = (const float*)d_in[14];
  float* out = (float*)d_out;

  char* cur = (char*)d_ws;
  float* hlin   = (float*)wsalloc(cur, (size_t)NN * 128 * 4);
  float* hagg   = (float*)wsalloc(cur, (size_t)NN * 128 * 4);
  float* xp1    = (float*)wsalloc(cur, (size_t)N2_ * 128 * 4);
  float* xp2    = (float*)wsalloc(cur, (size_t)N3_ * 128 * 4);
  float* dinvA  = (float*)wsalloc(cur, (size_t)NN * 4);
  float* dinvB  = (float*)wsalloc(cur, (size_t)N2_ * 4);
  float* scoreA = (float*)wsalloc(cur, (size_t)NN * 4);
  float* x1     = (float*)wsalloc(cur, (size_t)G_ * 256 * 4);
  float* x2     = (float*)wsalloc(cur, (size_t)G_ * 256 * 4);
  float* ew1    = (float*)wsalloc(cur, (size_t)EE * 4);
  int*   perm1  = (int*)  wsalloc(cur, (size_t)N2_ * 4);
  int*   perm2  = (int*)  wsalloc(cur, (size_t)N3_ * 4);
  int*   kept   = (int*)  wsalloc(cur, (size_t)NN * 4);
  int*   nmap   = (int*)  wsalloc(cur, (size_t)NN * 4);
  int*   kept2  = (int*)  wsalloc(cur, (size_t)N2_ * 4);
  int*   nmap2  = (int*)  wsalloc(cur, (size_t)N2_ * 4);
  int*   s1     = (int*)  wsalloc(cur, (size_t)EE * 4);
  int*   d1     = (int*)  wsalloc(cur, (size_t)EE * 4);

  const int T = 256;

  // ---------------- Layer 1: GCNConv + ReLU ----------------
  deg_dinv<NPG_><<<G_, NPG_, 0, stream>>>(dinvA, dst, nullptr);
  gemm128_wmma<<<NN / 128, 256, 0, stream>>>(x, W1, hlin);
  agg_relu_lds<NPG_><<<G_ * 2, 1024, 0, stream>>>(hagg, hlin, dinvA, b1, src, dst, nullptr);

  // ---------------- SAGPool 1 ----------------
  score_topk<NPG_, K1_><<<G_, NPG_, 0, stream>>>(scoreA, perm1, kept, nmap,
                                                 hagg, dinvA, Ws1, bs1, src, dst, nullptr);
  gather_tanh<<<cdiv(N2_ * 128, T), T, 0, stream>>>(xp1, hagg, scoreA, perm1, N2_);
  readout_k<K1_><<<G_, 128, 0, stream>>>(x1, xp1);
  remap_edges<<<cdiv(EE, T), T, 0, stream>>>(s1, d1, ew1, src, dst, kept, nmap, EE);

  // ---------------- Layer 2: GCNConv + ReLU ----------------
  deg_dinv<K1_><<<G_, K1_, 0, stream>>>(dinvB, d1, ew1);
  gemm128_wmma<<<N2_ / 128, 256, 0, stream>>>(xp1, W2, hlin);
  agg_relu_lds<K1_><<<G_ * 2, 1024, 0, stream>>>(hagg, hlin, dinvB, b2, s1, d1, ew1);

  // ---------------- SAGPool 2 (pooled edges unused downstream) ----------------
  score_topk<K1_, K2_><<<G_, K1_, 0, stream>>>(scoreA, perm2, kept2, nmap2,
                                               hagg, dinvB, Ws2, bs2, s1, d1, ew1);
  gather_tanh<<<cdiv(N3_ * 128, T), T, 0, stream>>>(xp2, hagg, scoreA, perm2, N3_);
  readout_k<K2_><<<G_, 128, 0, stream>>>(x2, xp2);

  // ---------------- Head ----------------
  mlp_k<<<G_, 128, 0, stream>>>(out, x1, x2, Wl1, bl1, Wl2, bl2);
}